// Specificity_66365834657894
// MI455X (gfx1250) — compile-verified
//
#include <hip/hip_runtime.h>

// Specificity metric for (1e6 x 100) logits.
// Bandwidth-bound: 400 MB of y_pred reads ~= 17 us at 23.3 TB/s; everything
// else (histograms, final 100-class math) is noise. Strategy:
//   K0: zero 300 u32 counters in d_ws (graphs replay: ws must be re-inited).
//   K1: wave-per-row argmax (lanes 0..24 load float4 -> global_load_b128,
//       wave32 shuffle reduce with lowest-index tie-break), LDS-privatized
//       histograms of y_true / argmax / diagonal, atomic flush to d_ws.
//       global_prefetch_b8 one wave-stride ahead (stride too large for HW).
//   K2: one wave computes per-class specificity and sums the 100 values with
//       two chained v_wmma_f32_16x16x4_f32 (A = ones => D row = column sums;
//       layout-permutation invariant, exact f32).

#define N_CLASSES 100
#define EPS 1e-7f

typedef __attribute__((ext_vector_type(2))) float v2f;
typedef __attribute__((ext_vector_type(4))) float v4f;
typedef __attribute__((ext_vector_type(8))) float v8f;

__global__ void spec_zero_kernel(unsigned* __restrict__ h) {
  int i = threadIdx.x;
  if (i < 3 * N_CLASSES) h[i] = 0u;
}

__global__ __launch_bounds__(256) void spec_hist_kernel(
    const float* __restrict__ y_pred, const int* __restrict__ y_true,
    unsigned* __restrict__ g_hist, int n_rows) {
  __shared__ unsigned s_hist[3 * N_CLASSES];
  for (int i = threadIdx.x; i < 3 * N_CLASSES; i += blockDim.x) s_hist[i] = 0u;
  __syncthreads();

  const int lane   = threadIdx.x & 31;
  const int wave   = (int)((blockIdx.x * blockDim.x + threadIdx.x) >> 5);
  const int nwaves = (int)((gridDim.x * blockDim.x) >> 5);

  for (int row = wave; row < n_rows; row += nwaves) {
    const float* rp = y_pred + (size_t)row * N_CLASSES;
    if (row + nwaves < n_rows)  // wave-uniform; stride ~nwaves*400B: must prefetch explicitly
      __builtin_prefetch(y_pred + (size_t)(row + nwaves) * N_CLASSES + lane * 4, 0, 1);

    float v  = -__builtin_inff();
    int  idx = 0x7fffffff;
    if (lane < 25) {  // 25 lanes x float4 = 100 contiguous floats (16B-aligned: 400B rows)
      v4f x = *(const v4f*)(rp + lane * 4);
      v = x.x; idx = lane * 4;
      if (x.y > v) { v = x.y; idx = lane * 4 + 1; }
      if (x.z > v) { v = x.z; idx = lane * 4 + 2; }
      if (x.w > v) { v = x.w; idx = lane * 4 + 3; }
    }
#pragma unroll
    for (int off = 16; off > 0; off >>= 1) {  // wave32 argmax, lowest index wins ties
      float ov = __shfl_down(v, off, 32);
      int   oi = __shfl_down(idx, off, 32);
      if (ov > v || (ov == v && oi < idx)) { v = ov; idx = oi; }
    }
    if (lane == 0) {
      int t = y_true[row];
      atomicAdd(&s_hist[t], 1u);                   // row_sum (actual-class marginal)
      atomicAdd(&s_hist[N_CLASSES + idx], 1u);     // col_sum (predicted-class marginal)
      if (t == idx) atomicAdd(&s_hist[2 * N_CLASSES + t], 1u);  // diagonal
    }
  }
  __syncthreads();
  for (int i = threadIdx.x; i < 3 * N_CLASSES; i += blockDim.x) {
    unsigned c = s_hist[i];
    if (c) atomicAdd(&g_hist[i], c);
  }
}

__device__ __forceinline__ float spec_of(const unsigned* __restrict__ h, int c,
                                         float total) {
  int cc = (c < N_CLASSES) ? c : (N_CLASSES - 1);   // clamp: no OOB even if speculated
  float rs = (float)h[cc];
  float cs = (float)h[N_CLASSES + cc];
  float dg = (float)h[2 * N_CLASSES + cc];
  float TN = total - rs - cs + dg;
  float FP = rs - dg;
  float s  = TN / (TN + FP + EPS);
  return (c < N_CLASSES) ? s : 0.0f;
}

__global__ void spec_final_kernel(const unsigned* __restrict__ h,
                                  float* __restrict__ out, int n_rows) {
  const int lane = threadIdx.x & 31;
  const float total = (float)n_rows;

  // Two B operands (4x16 f32 each = 64 slots) hold the 100 per-class values;
  // hardware B layout: lane<16 -> {K=0,K=1}, lane>=16 -> {K=2,K=3}, N=lane%16.
  // With A == ones the sum is layout-permutation invariant anyway.
  int c0 = (lane < 16) ? lane      : lane + 16;   // b covers classes 0..63
  int c1 = (lane < 16) ? lane + 16 : lane + 32;
  v2f a;  a.x = 1.0f; a.y = 1.0f;
  v2f b1; b1.x = spec_of(h, c0, total);      b1.y = spec_of(h, c1, total);
  v2f b2; b2.x = spec_of(h, c0 + 64, total); b2.y = spec_of(h, c1 + 64, total); // 64..99, rest 0

  v8f acc = {};
  acc = __builtin_amdgcn_wmma_f32_16x16x4_f32(false, a, false, b1, (short)0, acc, false, false);
  acc = __builtin_amdgcn_wmma_f32_16x16x4_f32(false, a, false, b2, (short)0, acc, false, false);

  // D VGPR0, lanes 0..15 = row M=0 -> each element is a column sum; sum 16 of them.
  float s = acc[0];
  s += __shfl_down(s, 8, 16);
  s += __shfl_down(s, 4, 16);
  s += __shfl_down(s, 2, 16);
  s += __shfl_down(s, 1, 16);
  if (lane == 0) out[0] = s * (1.0f / (float)N_CLASSES);
}

extern "C" void kernel_launch(void* const* d_in, const int* in_sizes, int n_in,
                              void* d_out, int out_size, void* d_ws, size_t ws_size,
                              hipStream_t stream) {
  const float* y_pred = (const float*)d_in[0];
  const int*   y_true = (const int*)d_in[1];
  const int n_rows = in_sizes[1];

  unsigned* g_hist = (unsigned*)d_ws;   // 3*100 u32 counters
  float*    out    = (float*)d_out;

  spec_zero_kernel<<<1, 512, 0, stream>>>(g_hist);

  int blocks = 2048;                    // 16384 waves, ~61 rows each
  if (blocks * 8 > n_rows) blocks = (n_rows + 7) / 8;
  if (blocks < 1) blocks = 1;
  spec_hist_kernel<<<blocks, 256, 0, stream>>>(y_pred, y_true, g_hist, n_rows);

  spec_final_kernel<<<1, 32, 0, stream>>>(g_hist, out, n_rows);
}